// Model_55121610277032
// MI455X (gfx1250) — compile-verified
//
#include <hip/hip_runtime.h>

typedef float v2f __attribute__((ext_vector_type(2)));
typedef float v8f __attribute__((ext_vector_type(8)));
typedef int   v4i __attribute__((ext_vector_type(4)));
typedef __attribute__((address_space(1))) v4i gv4i;   // global 16B chunk
typedef __attribute__((address_space(3))) v4i lv4i;   // LDS 16B chunk

#define M_TOK  2048   // b*t
#define DIM    1024
#define NHEADS 4
#define NKEYS  128
#define DHALF  128
#define TOPK   32

// ---------------------------------------------------------------------------
// Async global->LDS helpers (CDNA5 GLOBAL_LOAD_ASYNC_TO_LDS_B128, ASYNCcnt)
// ---------------------------------------------------------------------------
#if __has_builtin(__builtin_amdgcn_global_load_async_to_lds_b128)
#define HAS_ASYNC_LDS 1
#else
#define HAS_ASYNC_LDS 0
#endif

__device__ __forceinline__ void async_copy16(const float* g, float* l)
{
#if HAS_ASYNC_LDS
    __builtin_amdgcn_global_load_async_to_lds_b128((gv4i*)g, (lv4i*)l, 0, 0);
#else
    *(float4*)l = *(const float4*)g;
#endif
}

__device__ __forceinline__ void wait_async_copies()
{
#if HAS_ASYNC_LDS
    asm volatile("s_wait_asynccnt 0" ::: "memory");
#endif
}

// ---------------------------------------------------------------------------
// Kernel 1: Q = X @ Wq   (2048x1024) = (2048x1024)(1024x1024), f32 WMMA.
// 128 threads = 4 waves; 64x64 C tile per block; K in 64-deep chunks,
// double-buffered in LDS with async copies overlapping WMMA compute.
// ---------------------------------------------------------------------------
#define TM 64
#define TN 64
#define TK 64
#define LPAD 12            // row stride 76 dwords: conflict-free A b64 + B b32 frags
#define GEMM_THREADS 128
#define NCHUNK (DIM / TK)  // 16

__global__ __launch_bounds__(GEMM_THREADS)
void qgemm_kernel(const float* __restrict__ X, const float* __restrict__ W,
                  float* __restrict__ Q)
{
    __shared__ float As[2][TM][TK + LPAD];   // A tile, [m][k]
    __shared__ float Bs[2][TK][TN + LPAD];   // W tile, [k][n] (natural layout)

    const int tid  = threadIdx.x;
    const int lane = tid & 31;
    const int wave = tid >> 5;

    const int m0 = blockIdx.y * TM;
    const int n0 = blockIdx.x * TN;

    const int r16  = lane & 15;        // M row within strip / N col within subtile
    const int koff = (lane >> 4) * 2;  // lanes 0-15 -> K+0/1, lanes 16-31 -> K+2/3
    const int mb   = wave * 16;

    v8f acc0 = {}, acc1 = {}, acc2 = {}, acc3 = {};

    // ---- prologue: stage chunk 0 into buffer 0 ----
    #pragma unroll
    for (int i = 0; i < 8; ++i) {
        int idx = tid + GEMM_THREADS * i;     // 0..1023
        int r   = idx >> 4;                   // 0..63
        int c4  = (idx & 15) * 4;             // 0..60
        async_copy16(X + (size_t)(m0 + r) * DIM + c4,       &As[0][r][c4]);
        async_copy16(W + (size_t)r        * DIM + n0 + c4,  &Bs[0][r][c4]);
    }
    wait_async_copies();
    __syncthreads();

    for (int c = 0; c < NCHUNK; ++c) {
        const int buf = c & 1;

        // ---- issue async stage of chunk c+1 into the other buffer ----
        if (c + 1 < NCHUNK) {
            const int k0 = (c + 1) * TK;
            #pragma unroll
            for (int i = 0; i < 8; ++i) {
                int idx = tid + GEMM_THREADS * i;
                int r   = idx >> 4;
                int c4  = (idx & 15) * 4;
                async_copy16(X + (size_t)(m0 + r) * DIM + k0 + c4,      &As[buf ^ 1][r][c4]);
                async_copy16(W + (size_t)(k0 + r) * DIM + n0 + c4,      &Bs[buf ^ 1][r][c4]);
            }
        }

        // ---- WMMA compute on current buffer (overlaps the async copies) ----
        #pragma unroll
        for (int kk = 0; kk < TK; kk += 4) {
            v2f a, b0, b1, b2, b3;
            a.x  = As[buf][mb + r16][kk + koff];
            a.y  = As[buf][mb + r16][kk + koff + 1];
            b0.x = Bs[buf][kk + koff][ 0 + r16];  b0.y = Bs[buf][kk + koff + 1][ 0 + r16];
            b1.x = Bs[buf][kk + koff][16 + r16];  b1.y = Bs[buf][kk + koff + 1][16 + r16];
            b2.x = Bs[buf][kk + koff][32 + r16];  b2.y = Bs[buf][kk + koff + 1][32 + r16];
            b3.x = Bs[buf][kk + koff][48 + r16];  b3.y = Bs[buf][kk + koff + 1][48 + r16];
            acc0 = __builtin_amdgcn_wmma_f32_16x16x4_f32(false, a, false, b0, (short)0, acc0, false, false);
            acc1 = __builtin_amdgcn_wmma_f32_16x16x4_f32(false, a, false, b1, (short)0, acc1, false, false);
            acc2 = __builtin_amdgcn_wmma_f32_16x16x4_f32(false, a, false, b2, (short)0, acc2, false, false);
            acc3 = __builtin_amdgcn_wmma_f32_16x16x4_f32(false, a, false, b3, (short)0, acc3, false, false);
        }

        wait_async_copies();
        __syncthreads();
    }

    // C/D layout: lanes 0-15 hold M=0..7 in v0..7 (N=lane), lanes 16-31 hold M=8..15.
    const int gm0 = m0 + wave * 16 + (lane >> 4) * 8;
    const int gn  = n0 + (lane & 15);
    #pragma unroll
    for (int v = 0; v < 8; ++v) {
        Q[(size_t)(gm0 + v) * DIM + gn +  0] = acc0[v];
        Q[(size_t)(gm0 + v) * DIM + gn + 16] = acc1[v];
        Q[(size_t)(gm0 + v) * DIM + gn + 32] = acc2[v];
        Q[(size_t)(gm0 + v) * DIM + gn + 48] = acc3[v];
    }
}

// ---------------------------------------------------------------------------
// Kernel 2: per-token dots -> two-stage top-32 (rank-by-count) -> softmax ->
// weighted gather of values rows. One 128-thread block per token.
// ---------------------------------------------------------------------------
__global__ __launch_bounds__(128)
void topk_attn_kernel(const float* __restrict__ Q, const float* __restrict__ K,
                      const float* __restrict__ V, float* __restrict__ Out)
{
    __shared__ float qs[DIM];
    __shared__ float dots[8][NKEYS];
    __shared__ float s1[NHEADS][TOPK], s2[NHEADS][TOPK];
    __shared__ int   i1[NHEADS][TOPK], i2[NHEADS][TOPK];
    __shared__ float cval[NHEADS][128];
    __shared__ int   cidx[NHEADS][128];
    __shared__ float fs2[128];
    __shared__ int   fvi[128];
    __shared__ float red[128];
    __shared__ float attnw[128];

    const int bt  = blockIdx.x;
    const int tid = threadIdx.x;

    // ---- load q row (1024 f32) ----
    const float* qrow = Q + (size_t)bt * DIM;
    #pragma unroll
    for (int i = 0; i < 8; ++i) qs[tid + 128 * i] = qrow[tid + 128 * i];
    __syncthreads();

    // ---- dots[h*2+p][n] = <q[h,p,:], keys[h,n,p,:]> ----
    #pragma unroll
    for (int j = 0; j < 8; ++j) {
        int pi = tid + 128 * j;           // 0..1023
        int hp = pi >> 7;                 // 0..7
        int n  = pi & 127;
        int h  = hp >> 1, p = hp & 1;
        const float* qv = &qs[h * 256 + p * 128];
        const float* kv = K + (size_t)((h * NKEYS + n) * 2 + p) * DHALF;
        float acc = 0.f;
        #pragma unroll 4
        for (int d = 0; d < DHALF; d += 4) {
            float4 k4 = *(const float4*)(kv + d);
            acc += qv[d] * k4.x + qv[d + 1] * k4.y + qv[d + 2] * k4.z + qv[d + 3] * k4.w;
        }
        dots[hp][n] = acc;
    }
    __syncthreads();

    // ---- stage 1: top-32 of 128 per (h,p) via rank counting (gives sorted order) ----
    for (int hp = 0; hp < 8; ++hp) {
        float v = dots[hp][tid];
        int rank = 0;
        for (int m = 0; m < NKEYS; ++m) {
            float o = dots[hp][m];
            rank += (o > v) || (o == v && m < tid);
        }
        if (rank < TOPK) {
            int h = hp >> 1;
            if ((hp & 1) == 0) { s1[h][rank] = v; i1[h][rank] = tid; }
            else               { s2[h][rank] = v; i2[h][rank] = tid; }
        }
    }
    __syncthreads();

    // ---- stage 2: top-32 of pair sums. s1/s2 sorted desc, so any top-32 pair
    // has (i+1)*(j+1) <= 32  ->  only 119 candidates. tid = candidate slot. ----
    {
        int ci = 0, cj = 0;
        bool valid = (tid < 119);
        if (valid) {
            int rem = tid;
            int i = 0;
            while (true) {
                int cnt = TOPK / (i + 1);
                if (rem < cnt) { ci = i; cj = rem; break; }
                rem -= cnt; ++i;
            }
        }
        #pragma unroll
        for (int h = 0; h < NHEADS; ++h) {
            if (valid) {
                cval[h][tid] = s1[h][ci] + s2[h][cj];
                cidx[h][tid] = i1[h][ci] * NKEYS + i2[h][cj];
            } else {
                cval[h][tid] = -__builtin_inff();
                cidx[h][tid] = 0;
            }
        }
    }
    __syncthreads();

    for (int h = 0; h < NHEADS; ++h) {
        float v = cval[h][tid];
        int rank = 0;
        for (int m = 0; m < 128; ++m) {
            float o = cval[h][m];
            rank += (o > v) || (o == v && m < tid);
        }
        if (rank < TOPK && tid < 119) {
            fs2[h * TOPK + rank] = v;
            fvi[h * TOPK + rank] = cidx[h][tid];
        }
    }
    __syncthreads();

    // ---- softmax over the combined 128 scores ----
    red[tid] = fs2[tid];
    __syncthreads();
    for (int s = 64; s > 0; s >>= 1) {
        if (tid < s) red[tid] = fmaxf(red[tid], red[tid + s]);
        __syncthreads();
    }
    float mx = red[0];
    __syncthreads();
    float e = expf(fs2[tid] - mx);
    red[tid] = e;
    __syncthreads();
    for (int s = 64; s > 0; s >>= 1) {
        if (tid < s) red[tid] += red[tid + s];
        __syncthreads();
    }
    attnw[tid] = e / red[0];
    __syncthreads();

    // ---- out[d] = sum_k attn[k] * values[vi[k]][d]; values rows are L2-hot ----
    float4 acc_a = {0.f, 0.f, 0.f, 0.f};
    float4 acc_b = {0.f, 0.f, 0.f, 0.f};
    for (int k = 0; k < 128; ++k) {
        if (k + 4 < 128)
            __builtin_prefetch((const void*)(V + (size_t)fvi[k + 4] * DIM + tid * 8), 0, 1);
        float w = attnw[k];
        const float4* row = (const float4*)(V + (size_t)fvi[k] * DIM);
        float4 va = row[tid];        // cols [4*tid, 4*tid+4)    — coalesced
        float4 vb = row[128 + tid];  // cols [512+4*tid, ...)
        acc_a.x += w * va.x; acc_a.y += w * va.y; acc_a.z += w * va.z; acc_a.w += w * va.w;
        acc_b.x += w * vb.x; acc_b.y += w * vb.y; acc_b.z += w * vb.z; acc_b.w += w * vb.w;
    }
    float* orow = Out + (size_t)bt * DIM;
    *(float4*)(orow + 4 * tid)       = acc_a;
    *(float4*)(orow + 512 + 4 * tid) = acc_b;
}

// ---------------------------------------------------------------------------
extern "C" void kernel_launch(void* const* d_in, const int* in_sizes, int n_in,
                              void* d_out, int out_size, void* d_ws, size_t ws_size,
                              hipStream_t stream)
{
    const float* x      = (const float*)d_in[0];  // (2,1024,1024)
    const float* w_q    = (const float*)d_in[1];  // (1024,1024)
    const float* keys   = (const float*)d_in[2];  // (4,128,2,128)
    const float* values = (const float*)d_in[3];  // (16384,1024)
    float*       out    = (float*)d_out;          // (2,1024,1024)
    float*       qbuf   = (float*)d_ws;           // 2048*1024 f32 = 8 MB scratch

    dim3 g1(DIM / TN, M_TOK / TM);   // (16, 32)
    qgemm_kernel<<<g1, GEMM_THREADS, 0, stream>>>(x, w_q, qbuf);
    topk_attn_kernel<<<M_TOK, 128, 0, stream>>>(qbuf, keys, values, out);
    (void)in_sizes; (void)n_in; (void)out_size; (void)ws_size;
}